// KnowledgeCircuit_45526653337604
// MI455X (gfx1250) — compile-verified
//
#include <hip/hip_runtime.h>

typedef __attribute__((ext_vector_type(2))) float        v2f;
typedef __attribute__((ext_vector_type(4))) float        v4f;
typedef __attribute__((ext_vector_type(8))) float        v8f;
typedef __attribute__((ext_vector_type(4))) unsigned int v4u;
typedef __attribute__((ext_vector_type(8))) int          v8i;
typedef __attribute__((ext_vector_type(4))) int          v4i;

#define D_MODEL   512
#define NCAND     32
#define ROWSTRIDE 516   // 512 floats + TDM pad (2 dwords per 256-dword interval)

// One 64-thread block (2 waves) per token.
//  1) TDM gather-mode: 2 descriptors x 16 indexed rows (2KB each) -> LDS (padded layout).
//  2) Phase 1: act[n] = dot(x, cn[n]) via v_wmma_f32_16x16x4_f32, A from LDS (bank-conflict-free),
//     B = x broadcast across the 16 columns.
//  3) Phase 2: out = sum_n w[n]*cn[n] from LDS, coalesced stores.
__global__ __launch_bounds__(64)
void knowledge_circuit_kernel(const float* __restrict__ x,
                              const float* __restrict__ know,
                              const float* __restrict__ gate,
                              const float* __restrict__ vmask,
                              const int*   __restrict__ cidx,
                              float* __restrict__ out)
{
    __shared__ float cn_sh[NCAND * ROWSTRIDE];   // 66048 B, TDM-padded rows
    __shared__ float x_sh[D_MODEL];
    __shared__ float act_sh[NCAND];
    __shared__ float w_sh[NCAND];

    const int tok  = blockIdx.x;
    const int tid  = threadIdx.x;
    const int lane = tid & 31;
    const int wave = tid >> 5;

    // Stage x (512 f32 = 128 x b128) into LDS.
    {
        const v4f* xv = (const v4f*)(x + (size_t)tok * D_MODEL);
        v4f* xs = (v4f*)x_sh;
        xs[tid]      = xv[tid];
        xs[tid + 64] = xv[tid + 64];
    }

    // ---- TDM gather: 32 candidate rows -> LDS (wave 0 issues; wave-uniform values) ----
    if (wave == 0) {
        const uint64_t ga       = (uint64_t)(uintptr_t)know;
        const uint32_t lds_base = (uint32_t)(uintptr_t)cn_sh;   // LDS byte offset
        const int*     ip       = cidx + (size_t)tok * NCAND;
        #pragma unroll
        for (int half = 0; half < 2; ++half) {
            v4u g0;
            g0.x = 0x80000001u;                         // count=1 | gather_mode=1 | 16-bit indices
            g0.y = lds_base + (uint32_t)half * (16u * ROWSTRIDE * 4u);
            g0.z = (uint32_t)ga;                        // global_addr[31:0]
            g0.w = (uint32_t)((ga >> 32) & 0x1FFFFFFu)  // global_addr[56:32]
                 | 0x80000000u;                         // type=2 ("image")
            v8i g1;
            g1[0] = (2 << 16)                           // data_size = 4 bytes
                  | (1 << 20)                           // pad_enable
                  | (7 << 22)                           // pad_interval = 256 dwords
                  | (1 << 25);                          // pad_amount  = 2 dwords
            g1[1] = (int)(512u << 16);                  // tensor_dim0 = 512 (lo16)
            g1[2] = (int)(8192u << 16);                 // tensor_dim0 hi=0 | tensor_dim1 lo16 = 8192
            g1[3] = (int)(512u << 16);                  // tensor_dim1 hi=0 | tile_dim0 = 512
            g1[4] = 16;                                 // tile_dim1 = #valid indices
            g1[5] = 512;                                // tensor_dim0_stride lo32
            g1[6] = 0;
            g1[7] = 0;
            v4i g2, g3;                                 // row_index_0..7 / 8..15 (2 x u16 per dword)
            #pragma unroll
            for (int j = 0; j < 4; ++j) {
                const int i0 = half * 16 + j * 2;
                g2[j] = (ip[i0]     & 0xffff) | (ip[i0 + 1] << 16);
                g3[j] = (ip[i0 + 8] & 0xffff) | (ip[i0 + 9] << 16);
            }
            v8i g4 = {0, 0, 0, 0, 0, 0, 0, 0};          // unused extra group (zero-filled)
            __builtin_amdgcn_tensor_load_to_lds(g0, g1, g2, g3, g4, 0);
        }
        __builtin_amdgcn_s_wait_tensorcnt(0);
    }
    __syncthreads();

    // ---- Phase 1: act via WMMA f32 16x16x4, A/B K-pair = {koff, koff+1}, koff=2*(lane>=16) ----
    {
        const int m    = lane & 15;
        const int koff = (lane >> 4) << 1;
        const float* ap = cn_sh + (wave * 16 + m) * ROWSTRIDE + koff;
        const float* xp = x_sh + koff;

        v8f c = {};
        #pragma unroll 4
        for (int k0 = 0; k0 < 256; k0 += 4) {
            v2f a = *(const v2f*)(ap + k0);
            v2f b = *(const v2f*)(xp + k0);
            c = __builtin_amdgcn_wmma_f32_16x16x4_f32(false, a, false, b, (short)0, c, false, false);
        }
        ap += 2;                                        // skip mid-row TDM pad (2 dwords @ 256)
        #pragma unroll 4
        for (int k0 = 256; k0 < 512; k0 += 4) {
            v2f a = *(const v2f*)(ap + k0);
            v2f b = *(const v2f*)(xp + k0);
            c = __builtin_amdgcn_wmma_f32_16x16x4_f32(false, a, false, b, (short)0, c, false, false);
        }
        // C/D: VGPR r -> M=r (lanes 0-15) / M=r+8 (lanes 16-31); columns identical.
        if (lane == 0) {
            #pragma unroll
            for (int r = 0; r < 8; ++r) act_sh[wave * 16 + r] = c[r];
        }
        if (lane == 16) {
            #pragma unroll
            for (int r = 0; r < 8; ++r) act_sh[wave * 16 + 8 + r] = c[r];
        }
    }
    __syncthreads();

    // w[n] = act[n] * gate[n] * valid_mask[n]
    if (tid < NCAND) {
        const size_t gi = (size_t)tok * NCAND + tid;
        w_sh[tid] = act_sh[tid] * gate[gi] * vmask[gi];
    }
    __syncthreads();

    // ---- Phase 2: out[d] = sum_n w[n]*cn[n][d] from LDS (8B-aligned v2f, pad-aware) ----
    {
        v2f a0 = {}, a1 = {}, a2 = {}, a3 = {};
        const int d0 = tid * 4;                          // 0..252
        #pragma unroll 4
        for (int n = 0; n < NCAND; ++n) {
            const float  wn = w_sh[n];
            const float* rp = cn_sh + n * ROWSTRIDE;
            a0 += wn * *(const v2f*)(rp + d0);
            a1 += wn * *(const v2f*)(rp + d0 + 2);
            a2 += wn * *(const v2f*)(rp + d0 + 258);     // +256 data + 2 pad
            a3 += wn * *(const v2f*)(rp + d0 + 260);
        }
        float* op = out + (size_t)tok * D_MODEL;
        v4f o0 = { a0.x, a0.y, a1.x, a1.y };
        v4f o1 = { a2.x, a2.y, a3.x, a3.y };
        *(v4f*)(op + d0)       = o0;
        *(v4f*)(op + d0 + 256) = o1;
    }
}

extern "C" void kernel_launch(void* const* d_in, const int* in_sizes, int n_in,
                              void* d_out, int out_size, void* d_ws, size_t ws_size,
                              hipStream_t stream) {
    const float* x     = (const float*)d_in[0];  // (B,S,D)
    const float* know  = (const float*)d_in[1];  // (POOL,D)
    const float* gate  = (const float*)d_in[2];  // (B,S,N)
    const float* vmask = (const float*)d_in[3];  // (B,S,N)
    const int*   cidx  = (const int*)d_in[4];    // (B,S,N)
    float* out = (float*)d_out;

    const int tokens = in_sizes[0] / D_MODEL;    // B*S
    knowledge_circuit_kernel<<<tokens, 64, 0, stream>>>(x, know, gate, vmask, cidx, out);
}